// PetriNetAlignmentPredictor_71004399337496
// MI455X (gfx1250) — compile-verified
//
#include <hip/hip_runtime.h>
#include <hip/hip_bf16.h>
#include <math.h>

// ---------------------------------------------------------------------------
// PetriNetAlignmentPredictor for MI455X (gfx1250, wave32).
// expm(Omega)@v computed via truncated Taylor series (||Omega|| ~ 0.46).
// MLP GEMVs run on v_wmma_f32_16x16x4_f32 (f32-exact WMMA).
// Global->LDS staging uses GLOBAL_LOAD_ASYNC_TO_LDS_B64 (ASYNCcnt).
// ---------------------------------------------------------------------------

#define NUM_M     512
#define NUM_T     64
#define MAX_STEPS 20
#define OMEGA_ELEMS (NUM_M * NUM_M)   // 262144
#define NWG3      16                  // workgroups in the Taylor kernel
#define ROWS_PER_WG (NUM_M / NWG3)    // 32
#define NTERMS    14                  // Taylor terms (0.46^14/14! ~ 1e-16)

typedef __attribute__((ext_vector_type(2))) float v2f;
typedef __attribute__((ext_vector_type(8))) float v8f;
typedef __attribute__((ext_vector_type(2))) int   v2i;

// ---------------------------------------------------------------------------
// Async global->LDS copy (CDNA5 GLOBAL_LOAD_ASYNC_TO_LDS_B64, ASYNCcnt),
// guarded so the file compiles on toolchains without the builtins.
// Builtin signature (from hipcc diagnostic): (v2i*, v2i*, imm offset, imm cpol)
// ---------------------------------------------------------------------------
#if __has_builtin(__builtin_amdgcn_global_load_async_to_lds_b64)
#define HAS_ASYNC_LDS 1
#else
#define HAS_ASYNC_LDS 0
#endif

__device__ inline void async_copy_b64(const float* g, float* l) {
#if HAS_ASYNC_LDS
    __builtin_amdgcn_global_load_async_to_lds_b64((v2i*)(void*)g, (v2i*)(void*)l, 0, 0);
#else
    l[0] = g[0];
    l[1] = g[1];
#endif
}

__device__ inline void async_wait() {
#if HAS_ASYNC_LDS
#if __has_builtin(__builtin_amdgcn_s_wait_asynccnt)
    __builtin_amdgcn_s_wait_asynccnt(0);
#else
    asm volatile("s_wait_asynccnt 0x0" ::: "memory");
#endif
#endif
}

// ctrl layout (uint32): [0]=done, [1]=tgt_idx, [2..17]=barrier counters
// ---------------------------------------------------------------------------
// Inter-workgroup barrier: counter-per-instance, counters zeroed each step by
// the MLP kernel. Release fence before arrive, acquire fence after.
// ---------------------------------------------------------------------------
__device__ inline void gbar(unsigned* cnt, int slot) {
    __threadfence();
    __syncthreads();
    if (threadIdx.x == 0) {
        atomicAdd(&cnt[slot], 1u);
        while (__hip_atomic_load(&cnt[slot], __ATOMIC_RELAXED,
                                 __HIP_MEMORY_SCOPE_AGENT) < (unsigned)NWG3) {
            __builtin_amdgcn_s_sleep(1);
        }
    }
    __syncthreads();
    __threadfence();
}

// ---------------------------------------------------------------------------
// GEMV (1 x K) @ (K x N) via WMMA f32 16x16x4, M padded to 16 by BROADCASTING
// x into every M row (branchless A fragment: one ds_load_b64 per step).
// All 16 rows of D equal x@W; row M=0 is read from acc[0] of lanes 0-15.
// A layout (16x4 f32): lanes 0-15 hold {A[m][k0], A[m][k0+1]},
//                      lanes 16-31 hold {A[m][k0+2], A[m][k0+3]}.
// B layout (4x16 f32): VGPR0 = rows K0 (lanes 0-15) | K2 (lanes 16-31),
//                      VGPR1 = rows K1 | K3.
// One wave per 16-column output tile; two accumulators to pipeline the chain.
// ---------------------------------------------------------------------------
__device__ inline void gemv_wmma(const float* xin, const float* __restrict__ W,
                                 const float* __restrict__ bias, float* hout,
                                 int K, int N, bool do_relu, int wv, int lane) {
    int ntiles = N >> 4;
    if (wv >= ntiles) return;
    int n0  = wv << 4;
    int col = lane & 15;
    int kb  = (lane < 16) ? 0 : 2;
    v8f acc0 = {0.f,0.f,0.f,0.f,0.f,0.f,0.f,0.f};
    v8f acc1 = {0.f,0.f,0.f,0.f,0.f,0.f,0.f,0.f};
    #pragma unroll 2
    for (int k0 = 0; k0 < K; k0 += 8) {
        v2f a0;
        a0.x = xin[k0 + kb];
        a0.y = xin[k0 + kb + 1];
        v2f b0;
        b0.x = W[(k0 + kb) * N + n0 + col];
        b0.y = W[(k0 + kb + 1) * N + n0 + col];
        acc0 = __builtin_amdgcn_wmma_f32_16x16x4_f32(false, a0, false, b0,
                                                     (short)0, acc0, false, false);
        int k1 = k0 + 4;
        v2f a1;
        a1.x = xin[k1 + kb];
        a1.y = xin[k1 + kb + 1];
        v2f b1v;
        b1v.x = W[(k1 + kb) * N + n0 + col];
        b1v.y = W[(k1 + kb + 1) * N + n0 + col];
        acc1 = __builtin_amdgcn_wmma_f32_16x16x4_f32(false, a1, false, b1v,
                                                     (short)0, acc1, false, false);
    }
    if (lane < 16) {
        float h = acc0[0] + acc1[0] + bias[n0 + col];
        if (do_relu) h = fmaxf(h, 0.f);
        hout[n0 + col] = h;
    }
}

// ---------------------------------------------------------------------------
// Init: zero d_out, v <- v_src, ctrl cleared, tgt_idx = argmax(v_target)
// ---------------------------------------------------------------------------
__global__ __launch_bounds__(512) void k_init(const float* __restrict__ v_src,
                                              const float* __restrict__ v_tgt,
                                              float* __restrict__ v,
                                              float* __restrict__ out, int out_n,
                                              unsigned* __restrict__ ctrl) {
    int tid = threadIdx.x;
    for (int i = tid; i < out_n; i += blockDim.x) out[i] = 0.f;
    for (int i = tid; i < NUM_M; i += blockDim.x) v[i] = v_src[i];
    if (tid < 32) ctrl[tid] = 0u;
    __syncthreads();
    if (tid == 0) {
        float best = v_tgt[0]; unsigned bi = 0;
        for (int i = 1; i < NUM_M; ++i) {
            float x = v_tgt[i];
            if (x > best) { best = x; bi = (unsigned)i; }
        }
        ctrl[1] = bi;
    }
}

// ---------------------------------------------------------------------------
// Per-step: MLP (3 WMMA layers) + gumbel softmax + logits output.
// Single workgroup, 512 threads = 16 waves. Also resets barrier counters.
// ---------------------------------------------------------------------------
__global__ __launch_bounds__(512) void k_mlp(
    const float* __restrict__ v, const float* __restrict__ v_tgt,
    const float* __restrict__ W1, const float* __restrict__ b1,
    const float* __restrict__ W2, const float* __restrict__ b2,
    const float* __restrict__ W3, const float* __restrict__ b3,
    const float* __restrict__ gumbel,
    float* __restrict__ soft_out, float* __restrict__ out_logits,
    unsigned* __restrict__ ctrl, int step) {
    __shared__ float xs[2 * NUM_M];
    __shared__ float h1[256];
    __shared__ float h2[128];
    __shared__ float lg[NUM_T];
    __shared__ float zz[NUM_T];
    __shared__ float ee[NUM_T];
    __shared__ float red[2];
    __shared__ unsigned sdone;

    int tid = threadIdx.x;
    int lane = tid & 31;
    int wv = tid >> 5;

    if (tid < 16) ctrl[2 + tid] = 0u;          // reset barrier counters
    if (tid == 0) sdone = ctrl[0];
    // stage x = [v, v_target] into LDS via async copy (2 floats/thread)
    if (tid < 256) async_copy_b64(v + tid * 2, &xs[tid * 2]);
    else           async_copy_b64(v_tgt + (tid - 256) * 2, &xs[NUM_M + (tid - 256) * 2]);
    async_wait();
    __syncthreads();
    if (sdone) return;                          // frozen: logits row stays 0

    gemv_wmma(xs, W1, b1, h1, 2 * NUM_M, 256, true, wv, lane);
    __syncthreads();
    gemv_wmma(h1, W2, b2, h2, 256, 128, true, wv, lane);
    __syncthreads();
    gemv_wmma(h2, W3, b3, lg, 128, NUM_T, false, wv, lane);
    __syncthreads();

    // gumbel softmax, tau = 1
    if (tid < NUM_T) zz[tid] = lg[tid] + gumbel[step * NUM_T + tid];
    __syncthreads();
    if (tid == 0) {
        float m = zz[0];
        for (int t = 1; t < NUM_T; ++t) m = fmaxf(m, zz[t]);
        red[0] = m;
    }
    __syncthreads();
    if (tid < NUM_T) ee[tid] = expf(zz[tid] - red[0]);
    __syncthreads();
    if (tid == 0) {
        float s = 0.f;
        for (int t = 0; t < NUM_T; ++t) s += ee[t];
        red[1] = 1.f / s;
    }
    __syncthreads();
    if (tid < NUM_T) {
        soft_out[tid] = ee[tid] * red[1];
        out_logits[step * NUM_T + tid] = lg[tid];   // pre-gumbel logits
    }
}

// ---------------------------------------------------------------------------
// Omega = sum_t soft[t] * omegas[t]  -- pure bandwidth (64 MB/step, L2-resident)
// float4 per thread, device-wide grid, prefetch the t-slice two ahead.
// ---------------------------------------------------------------------------
__global__ __launch_bounds__(256) void k_omega(const float* __restrict__ omegas,
                                               const float* __restrict__ soft,
                                               float* __restrict__ omega,
                                               const unsigned* __restrict__ ctrl) {
    __shared__ float ss[NUM_T];
    __shared__ unsigned sdone;
    int tid = threadIdx.x;
    if (tid == 0) sdone = ctrl[0];
    if (tid < NUM_T) ss[tid] = soft[tid];
    __syncthreads();
    if (sdone) return;

    int idx4 = blockIdx.x * 256 + tid;          // float4 index
    const float4* op = (const float4*)omegas;
    float4 acc; acc.x = 0.f; acc.y = 0.f; acc.z = 0.f; acc.w = 0.f;
    #pragma unroll 4
    for (int t = 0; t < NUM_T; ++t) {
        if (t + 2 < NUM_T)
            __builtin_prefetch((const void*)(op + (size_t)(t + 2) * (OMEGA_ELEMS / 4) + idx4), 0, 0);
        float s = ss[t];
        float4 w = op[(size_t)t * (OMEGA_ELEMS / 4) + idx4];
        acc.x += s * w.x; acc.y += s * w.y; acc.z += s * w.z; acc.w += s * w.w;
    }
    ((float4*)omega)[idx4] = acc;
}

// ---------------------------------------------------------------------------
// v <- expm(Omega) @ v via Taylor: p_j = Omega p_{j-1} / j ; y = sum p_j.
// 16 workgroups x 256 threads (8 waves). Wave-per-row, b128 fully-coalesced
// row reads from L2; p staged in LDS (async) each term; inter-WG barrier per
// term. Then global argmax(v_new) -> done flag.
// ---------------------------------------------------------------------------
__global__ __launch_bounds__(256) void k_taylor(const float* __restrict__ A,
                                                float* __restrict__ v,
                                                float* __restrict__ pb0,
                                                float* __restrict__ pb1,
                                                unsigned* __restrict__ ctrl,
                                                float* __restrict__ avals,
                                                unsigned* __restrict__ aidx) {
    __shared__ float p[NUM_M];
    __shared__ float y[ROWS_PER_WG];
    __shared__ unsigned sdone;
    int tid = threadIdx.x;
    int lane = tid & 31;
    int wv = tid >> 5;                 // 0..7
    int wg = blockIdx.x;               // 0..15
    unsigned* cnt = ctrl + 2;

    if (tid == 0) sdone = ctrl[0];
    __syncthreads();
    if (sdone) return;                 // v frozen once done

    if (tid < ROWS_PER_WG) y[tid] = v[wg * ROWS_PER_WG + tid];   // j=0 term

    for (int j = 1; j <= NTERMS; ++j) {
        const float* src = (j == 1) ? v : (((j - 1) & 1) ? pb1 : pb0);
        float* dst = (j & 1) ? pb1 : pb0;
        async_copy_b64(src + tid * 2, &p[tid * 2]);   // 512 floats -> LDS
        async_wait();
        __syncthreads();
        float invj = 1.0f / (float)j;
        #pragma unroll
        for (int i = 0; i < ROWS_PER_WG / 8; ++i) {
            int rl = wv + 8 * i;
            int r = wg * ROWS_PER_WG + rl;
            const float4* Ar4 = (const float4*)(A + (size_t)r * NUM_M);
            float part = 0.f;
            #pragma unroll
            for (int c = 0; c < 4; ++c) {
                float4 av = Ar4[c * 32 + lane];
                float4 pv = *(const float4*)(&p[(c * 32 + lane) * 4]);
                part += av.x * pv.x + av.y * pv.y + av.z * pv.z + av.w * pv.w;
            }
            #pragma unroll
            for (int off = 16; off > 0; off >>= 1)
                part += __shfl_xor(part, off, 32);
            if (lane == 0) {
                float q = part * invj;
                y[rl] += q;
                dst[r] = q;
            }
        }
        gbar(cnt, j - 1);              // slots 0..NTERMS-1
    }

    // write v_new (not done -> always update, matching jnp.where)
    if (tid < ROWS_PER_WG) v[wg * ROWS_PER_WG + tid] = y[tid];
    // local argmax (first-max tiebreak, ascending scan)
    if (tid == 0) {
        float best = y[0]; unsigned bi = 0;
        for (int i = 1; i < ROWS_PER_WG; ++i)
            if (y[i] > best) { best = y[i]; bi = (unsigned)i; }
        avals[wg] = best;
        aidx[wg] = (unsigned)(wg * ROWS_PER_WG) + bi;
    }
    gbar(cnt, NTERMS);                 // slot NTERMS (< 16)
    if (wg == 0 && tid == 0) {
        float best = avals[0]; unsigned bi = aidx[0];
        for (int w = 1; w < NWG3; ++w)
            if (avals[w] > best) { best = avals[w]; bi = aidx[w]; }
        ctrl[0] = (bi == ctrl[1]) ? 1u : 0u;
    }
}

__global__ __launch_bounds__(512) void k_fin(const float* __restrict__ v,
                                             float* __restrict__ out) {
    int tid = threadIdx.x;
    if (tid < NUM_M) out[tid] = v[tid];
}

// ---------------------------------------------------------------------------
extern "C" void kernel_launch(void* const* d_in, const int* in_sizes, int n_in,
                              void* d_out, int out_size, void* d_ws, size_t ws_size,
                              hipStream_t stream) {
    const float* v_src  = (const float*)d_in[0];
    const float* v_tgt  = (const float*)d_in[1];
    const float* omegas = (const float*)d_in[2];
    const float* W1 = (const float*)d_in[3];
    const float* b1 = (const float*)d_in[4];
    const float* W2 = (const float*)d_in[5];
    const float* b2 = (const float*)d_in[6];
    const float* W3 = (const float*)d_in[7];
    const float* b3 = (const float*)d_in[8];
    const float* gum = (const float*)d_in[9];
    float* out = (float*)d_out;

    float* wsf = (float*)d_ws;
    float* v      = wsf;                         // 512
    float* soft   = wsf + 512;                   // 64
    float* pb0    = wsf + 576;                   // 512
    float* pb1    = wsf + 1088;                  // 512
    unsigned* ctrl = (unsigned*)(wsf + 1600);    // 32 dwords
    float* avals  = wsf + 1632;                  // 16
    unsigned* aidx = (unsigned*)(wsf + 1664);    // 16
    float* omega  = wsf + 2048;                  // 262144

    k_init<<<1, 512, 0, stream>>>(v_src, v_tgt, v, out, out_size, ctrl);
    for (int s = 0; s < MAX_STEPS; ++s) {
        k_mlp<<<1, 512, 0, stream>>>(v, v_tgt, W1, b1, W2, b2, W3, b3,
                                     gum, soft, out + NUM_M, ctrl, s);
        k_omega<<<OMEGA_ELEMS / 4 / 256, 256, 0, stream>>>(omegas, soft, omega, ctrl);
        k_taylor<<<NWG3, 256, 0, stream>>>(omega, v, pb0, pb1, ctrl, avals, aidx);
    }
    k_fin<<<1, 512, 0, stream>>>(v, out);
}